// DefocusLKPN_56873956933947
// MI455X (gfx1250) — compile-verified
//
#include <hip/hip_runtime.h>
#include <hip/hip_bf16.h>
#include <stdint.h>

#define KSZ   5
#define KKN   25
#define CH    4
#define HGT   128
#define WID   128
#define HW    (HGT * WID)
#define NB    16
#define UCH   101            // 4*25 logit channels + 1 delta channel
#define TILE  16
#define HALO  2
#define SW    (TILE + 2 * HALO)    // 20
#define SPC   (SW * SW)            // 400 floats per channel tile

typedef __attribute__((address_space(1))) int g_int;
typedef __attribute__((address_space(3))) int l_int;

// ---------------------------------------------------------------------------
// Bandwidth-bound elementwise/stencil kernel. One block = one 16x16 pixel
// tile of one image; x tile (+2 halo) for all 4 channels staged into LDS via
// async global->LDS copies; 101 unet_out channels streamed with NT hints.
// ---------------------------------------------------------------------------
__global__ __launch_bounds__(256)
void defocus_lkpn_kernel(const float* __restrict__ x,
                         const float* __restrict__ defocus,
                         const float* __restrict__ unet,
                         const float* __restrict__ alpha,
                         float* __restrict__ out)
{
    __shared__ float smem[CH * SPC];   // 6400 B

    const int b   = blockIdx.z;
    const int h0  = blockIdx.y * TILE;
    const int w0  = blockIdx.x * TILE;
    const int tid = threadIdx.x;

    // ---- stage x tile (with halo, zero padded) into LDS --------------------
    const float* xb = x + (size_t)b * (CH * HW);
    for (int i = tid; i < CH * SPC; i += 256) {
        const int c   = i / SPC;
        const int rem = i - c * SPC;
        const int r   = rem / SW;
        const int col = rem - r * SW;
        const int gh  = h0 - HALO + r;
        const int gw  = w0 - HALO + col;
        if ((unsigned)gh < (unsigned)HGT && (unsigned)gw < (unsigned)WID) {
#if __has_builtin(__builtin_amdgcn_global_load_async_to_lds_b32)
            __builtin_amdgcn_global_load_async_to_lds_b32(
                (g_int*)(xb + (size_t)c * HW + gh * WID + gw),
                (l_int*)(&smem[i]),
                0 /*offset*/, 0 /*cpol*/);
#else
            smem[i] = xb[(size_t)c * HW + gh * WID + gw];
#endif
        } else {
            smem[i] = 0.0f;
        }
    }
#if __has_builtin(__builtin_amdgcn_s_wait_asynccnt)
    __builtin_amdgcn_s_wait_asynccnt(0);
#else
    asm volatile("s_wait_asynccnt 0" ::: "memory");
#endif
    __syncthreads();

    // ---- per-pixel mask (shared across channels) ---------------------------
    const int ly = tid >> 4;
    const int lx = tid & 15;
    const int hw = (h0 + ly) * WID + (w0 + lx);

    const float dist_tab[KKN] = {
        2.8284271f, 2.2360680f, 2.0f,       2.2360680f, 2.8284271f,
        2.2360680f, 1.4142135f, 1.0f,       1.4142135f, 2.2360680f,
        2.0f,       1.0f,       0.0f,       1.0f,       2.0f,
        2.2360680f, 1.4142135f, 1.0f,       1.4142135f, 2.2360680f,
        2.8284271f, 2.2360680f, 2.0f,       2.2360680f, 2.8284271f };

    const float alphav = alpha[0];
    const float* ub = unet + (size_t)b * UCH * HW + hw;

    // delta_r = tanh(unet[:,100]) via exp (single fast transcendental)
    const float din = __builtin_nontemporal_load(ub + 100 * HW);
    const float e2  = __expf(2.0f * din);
    const float delta_r = (e2 - 1.0f) * __builtin_amdgcn_rcpf(e2 + 1.0f);

    float radius = alphav * __builtin_nontemporal_load(defocus + (size_t)b * HW + hw) + delta_r;
    radius = fminf(fmaxf(radius, 0.0f), 3.0f);

    float mask[KKN];
#pragma unroll
    for (int kk = 0; kk < KKN; ++kk) {
        const float t = 5.0f * (radius - dist_tab[kk]);       // BETA = 5
        mask[kk] = __builtin_amdgcn_rcpf(1.0f + __expf(-t));  // sigmoid
    }

    // ---- per-channel: softmax * mask, renormalize, 25-tap dot --------------
    float* outb = out + (size_t)b * CH * HW + hw;
    for (int c = 0; c < CH; ++c) {
        const float* lp = ub + (size_t)c * (KKN * HW);
        if (c + 1 < CH)
            __builtin_prefetch(ub + (size_t)(c + 1) * (KKN * HW), 0, 0);

        float l[KKN];
        float mx = -3.0e38f;
#pragma unroll
        for (int kk = 0; kk < KKN; ++kk)
            l[kk] = __builtin_nontemporal_load(lp + kk * HW);
#pragma unroll
        for (int kk = 0; kk < KKN; ++kk)
            mx = fmaxf(mx, l[kk]);

        float E = 0.0f, S = 0.0f;
#pragma unroll
        for (int kk = 0; kk < KKN; ++kk) {
            const float e = __expf(l[kk] - mx);   // softmax numerator
            E += e;
            const float n = e * mask[kk];         // masked weight (unnormalized)
            S += n;
            l[kk] = n;
        }
        // mk = n/E / max(S/E, eps)  ==  n / max(S, eps*E)
        const float rden = __builtin_amdgcn_rcpf(fmaxf(S, 1e-6f * E));

        const float* sc = &smem[c * SPC];
        float acc = 0.0f;
#pragma unroll
        for (int ky = 0; ky < KSZ; ++ky) {
#pragma unroll
            for (int kx = 0; kx < KSZ; ++kx) {
                acc = fmaf(l[ky * KSZ + kx], sc[(ly + ky) * SW + (lx + kx)], acc);
            }
        }
        const float xc = sc[(ly + HALO) * SW + (lx + HALO)];
        __builtin_nontemporal_store(acc * rden + xc, outb + (size_t)c * HW);
    }
}

extern "C" void kernel_launch(void* const* d_in, const int* in_sizes, int n_in,
                              void* d_out, int out_size, void* d_ws, size_t ws_size,
                              hipStream_t stream)
{
    (void)in_sizes; (void)n_in; (void)out_size; (void)d_ws; (void)ws_size;
    const float* x       = (const float*)d_in[0];
    const float* defocus = (const float*)d_in[1];
    const float* unet    = (const float*)d_in[2];
    const float* alpha   = (const float*)d_in[3];
    float*       out     = (float*)d_out;

    dim3 grid(WID / TILE, HGT / TILE, NB);   // 8 x 8 x 16 = 1024 blocks
    dim3 block(256);                         // 8 wave32 per block
    defocus_lkpn_kernel<<<grid, block, 0, stream>>>(x, defocus, unet, alpha, out);
}